// ConditionalRotatEHead_10539849744619
// MI455X (gfx1250) — compile-verified
//
#include <hip/hip_runtime.h>
#include <stdint.h>

// ConditionalRotatEHead scoring for MI455X (gfx1250, wave32).
// Gather-bound: ~520MB of node-row gathers; node table (51MB) is L2-resident
// (192MB L2). One edge per wave32; h/t rows staged into LDS with CDNA5 async
// global->LDS copies (ASYNCcnt), depth-2 software pipeline over 3 slots.

#define EMB   128
#define HALF  64
#define NREL  16
#define WPB   8            // waves per block
#define BLKT  256          // threads per block (8 x wave32)
#define NSLOT 3            // pipeline slots per wave (depth-2)

// ---------- CDNA5 async global->LDS (gfx1250) ----------
// One wave copies a 512B row: 32 lanes x b128. GVS addressing:
// reads at the row base in the SGPR pair plus the byte offset in vaddr, writes to the LDS address in vdst.
__device__ __forceinline__ void async_row_to_lds(const float* gbase,
                                                 uint32_t lds_byte,
                                                 uint32_t lane) {
  uint32_t voff = lane * 16u;
  uint32_t ldst = lds_byte + voff;
  asm volatile("global_load_async_to_lds_b128 %0, %1, %2"
               :: "v"(ldst), "v"(voff), "s"(gbase)
               : "memory");
}
// ASYNCcnt decrements in order for async loads -> waiting <=2 guarantees the
// oldest edge's two row-copies (h,t) have fully landed in LDS.
__device__ __forceinline__ void wait_async_le2() {
  asm volatile("s_wait_asynccnt 2" ::: "memory");
}
__device__ __forceinline__ void wait_async_0() {
  asm volatile("s_wait_asynccnt 0" ::: "memory");
}
__device__ __forceinline__ uint32_t lds_off(const void* p) {
  // the low 32 bits of a generic shared pointer are its LDS offset
  return (uint32_t)(uintptr_t)p;
}

// ---------- kernel 1: 16x64 cos/sin tables (precise, once) ----------
__global__ void rel_table_kernel(const float* __restrict__ rel_emb,
                                 float* __restrict__ cos_tbl,
                                 float* __restrict__ sin_tbl) {
  int i = threadIdx.x;                 // 1024 = NREL*HALF
  float s, c;
  sincosf(rel_emb[i], &s, &c);
  cos_tbl[i] = c;
  sin_tbl[i] = s;
}

// ---------- kernel 2: edge scoring ----------
__global__ __launch_bounds__(BLKT) void edge_score_kernel(
    const float* __restrict__ emb,
    const float* __restrict__ cos_tbl,
    const float* __restrict__ sin_tbl,
    const int*   __restrict__ eidx,    // [2, n_edges]
    const int*   __restrict__ rel,     // [n_edges]
    float*       __restrict__ out,
    int n_edges) {
  __shared__ __align__(16) float s_cos[NREL * HALF];                  // 4KB
  __shared__ __align__(16) float s_sin[NREL * HALF];                  // 4KB
  __shared__ __align__(16) float s_buf[WPB][NSLOT][2 * EMB];          // 24KB

  for (int i = threadIdx.x; i < NREL * HALF; i += BLKT) {
    s_cos[i] = cos_tbl[i];
    s_sin[i] = sin_tbl[i];
  }
  __syncthreads();

  const int lane   = threadIdx.x & 31;
  const int wave   = threadIdx.x >> 5;
  const int stride = gridDim.x * WPB;

  // Issue async h/t row copies for edge ei into slot; return its relation id.
  auto prefetch = [&](int ei, int slot) -> int {
    int hd = __builtin_amdgcn_readfirstlane(eidx[ei]);
    int tl = __builtin_amdgcn_readfirstlane(eidx[n_edges + ei]);
    int rt = __builtin_amdgcn_readfirstlane(rel[ei]);
    const float* hrow = emb + (size_t)hd * EMB;
    const float* trow = emb + (size_t)tl * EMB;
    uint32_t base = lds_off(&s_buf[wave][slot][0]);
    async_row_to_lds(hrow, base, (uint32_t)lane);
    async_row_to_lds(trow, base + 512u, (uint32_t)lane);
    return rt;
  };

  // Score edge ei from staged LDS slot.
  auto score = [&](int ei, int slot, int rt) {
    const float* hS = &s_buf[wave][slot][0];
    const float* tS = &s_buf[wave][slot][EMB];
    float2 hre = *(const float2*)(hS + 2 * lane);
    float2 him = *(const float2*)(hS + HALF + 2 * lane);
    float2 tre = *(const float2*)(tS + 2 * lane);
    float2 tim = *(const float2*)(tS + HALF + 2 * lane);

    // per-edge norms (recomputed from staged rows; ALU is free here)
    float sh = hre.x * hre.x + hre.y * hre.y + him.x * him.x + him.y * him.y;
    float st = tre.x * tre.x + tre.y * tre.y + tim.x * tim.x + tim.y * tim.y;
#pragma unroll
    for (int off = 16; off > 0; off >>= 1) {
      sh += __shfl_xor(sh, off, 32);
      st += __shfl_xor(st, off, 32);
    }
    float ih = 1.0f / fmaxf(sqrtf(sh), 1e-12f);
    float it = 1.0f / fmaxf(sqrtf(st), 1e-12f);
    hre.x *= ih; hre.y *= ih; him.x *= ih; him.y *= ih;
    tre.x *= it; tre.y *= it; tim.x *= it; tim.y *= it;

    float dot = hre.x * tre.x + hre.y * tre.y + him.x * tim.x + him.y * tim.y;

    float2 c = *(const float2*)(s_cos + rt * HALF + 2 * lane);
    float2 s = *(const float2*)(s_sin + rt * HALF + 2 * lane);
    float dre0 = hre.x * c.x - him.x * s.x - tre.x;
    float dim0 = hre.x * s.x + him.x * c.x - tim.x;
    float dre1 = hre.y * c.y - him.y * s.y - tre.y;
    float dim1 = hre.y * s.y + him.y * c.y - tim.y;
    float dsq = dre0 * dre0 + dim0 * dim0 + dre1 * dre1 + dim1 * dim1;

#pragma unroll
    for (int off = 16; off > 0; off >>= 1) {
      dot += __shfl_xor(dot, off, 32);
      dsq += __shfl_xor(dsq, off, 32);
    }

    if (lane == 0) {
      float dscore = dot * (1.0f / (float)EMB) - 1.0f;  // -(1 - mean)
      float rscore = -sqrtf(dsq + 1e-8f);
      bool  sym    = (rt <= 6) && ((rt & 1) == 0);      // {0,2,4,6}
      out[ei] = sym ? dscore : rscore;
    }
  };

  // Depth-2 pipeline over 3 slots: up to 4 async row-copies in flight.
  int e0 = blockIdx.x * WPB + wave;
  int e1 = e0 + stride;
  int rt0 = 0, rt1 = 0;
  if (e0 < n_edges) rt0 = prefetch(e0, 0);
  if (e1 < n_edges) rt1 = prefetch(e1, 1);

  int slot = 0;
  while (e0 < n_edges) {
    int e2 = e1 + stride;
    if (e1 < n_edges) wait_async_le2();   // oldest edge (e0) landed
    else              wait_async_0();     // tail: nothing behind e0

    int slot2 = slot + 2; if (slot2 >= NSLOT) slot2 -= NSLOT;
    int rt2 = 0;
    if (e2 < n_edges) rt2 = prefetch(e2, slot2);

    score(e0, slot, rt0);

    rt0 = rt1; rt1 = rt2;
    e0 = e1;   e1 = e2;
    slot = slot + 1; if (slot >= NSLOT) slot -= NSLOT;
  }
}

extern "C" void kernel_launch(void* const* d_in, const int* in_sizes, int n_in,
                              void* d_out, int out_size, void* d_ws, size_t ws_size,
                              hipStream_t stream) {
  const float* emb     = (const float*)d_in[0];
  const float* rel_emb = (const float*)d_in[1];
  const int*   eidx    = (const int*)d_in[2];
  const int*   rel     = (const int*)d_in[3];
  float*       out     = (float*)d_out;
  const int n_edges = in_sizes[3];

  float* cos_tbl = (float*)d_ws;              // 4KB
  float* sin_tbl = cos_tbl + NREL * HALF;     // 4KB

  rel_table_kernel<<<1, NREL * HALF, 0, stream>>>(rel_emb, cos_tbl, sin_tbl);

  int blocks = (n_edges + WPB - 1) / WPB;
  if (blocks > 2048) blocks = 2048;           // grid-stride amortizes LDS table
  edge_score_kernel<<<blocks, BLKT, 0, stream>>>(emb, cos_tbl, sin_tbl,
                                                 eidx, rel, out, n_edges);
}